// ConditionalFilterLayer_58385785422118
// MI455X (gfx1250) — compile-verified
//
#include <hip/hip_runtime.h>
#include <hip/hip_bf16.h>

// ---------------------------------------------------------------------------
// ConditionalFilterLayer, MI455X (gfx1250), f32 via V_WMMA_F32_16X16X4_F32.
// Memory-bound (x = 268MB > 192MB L2): 2 HBM passes over x total.
//   K1: fused mask-conv + sigmoid + masked pooling  (pass 1 over x)
//   K2: grouped per-class 512x512 GEMM on pooled features (tiny)
//   K3: pred = filters . x                           (pass 2 over x)
// ---------------------------------------------------------------------------

typedef __attribute__((ext_vector_type(2))) float v2f;
typedef __attribute__((ext_vector_type(8))) float v8f;

#define B_DIM   8
#define C_DIM   512
#define K_CLS   19
#define KP      32            // k padded to 2 WMMA M-tiles
#define S_DIM   16384         // h*w
#define TS      128           // spatial tile per block
#define NTHR    256           // 8 waves (wave32)
#define WPITCH  516           // 512 + 4: row pitch for weight tiles (bank-conflict-free)
#define XPITCH  132           // 128 + 4: row pitch for x tiles
#define MPITCH  132           // 128 + 4: row pitch for mask tile

__device__ __forceinline__ v8f wmma_f32(v2f a, v2f b, v8f c) {
  // D = A(16x4) * B(4x16) + C(16x16), all f32 (exact vs reference)
  return __builtin_amdgcn_wmma_f32_16x16x4_f32(false, a, false, b, (short)0, c,
                                               false, false);
}

// ---------------------------------------------------------------------------
// K1: per (batch, 128-wide spatial tile):
//   GEMM1: pre_mask[32,128] = Wm_pad[32,512] . x[512,128]   (WMMA, K=c)
//   sigmoid(+bias) -> mask in LDS
//   GEMM2: cf[32,512]      += mask[32,128]   . xT[128,512]  (WMMA, K=s)
// x chunks staged through LDS twice (2nd stage hits L2).
// ---------------------------------------------------------------------------
__global__ void __launch_bounds__(NTHR)
k1_mask_pool(const float* __restrict__ x, const float* __restrict__ Wm,
             const float* __restrict__ bm, float* __restrict__ cf) {
  extern __shared__ float lds[];
  float* lW = lds;                    // KP * WPITCH   (Wm zero-padded)
  float* lX = lW + KP * WPITCH;       // TS * XPITCH   (x chunk: 128 c rows x 128 s)
  float* lM = lX + TS * XPITCH;       // KP * MPITCH   (mask tile)
  float* lB = lM + KP * MPITCH;       // KP            (bias zero-padded)

  const int t  = threadIdx.x;
  const int bi = blockIdx.y;
  const int s0 = blockIdx.x * TS;

  // stage Wm [19,512] zero-padded into lW [32,516]
  for (int i = t; i < KP * WPITCH; i += NTHR) lW[i] = 0.0f;
  __syncthreads();
  for (int q = t; q < (K_CLS * C_DIM) / 4; q += NTHR) {
    const int row = q / (C_DIM / 4);
    const int col = (q % (C_DIM / 4)) * 4;
    const float4 v = *(const float4*)&Wm[row * C_DIM + col];
    float* d = &lW[row * WPITCH + col];
    d[0] = v.x; d[1] = v.y; d[2] = v.z; d[3] = v.w;
  }
  if (t < KP) lB[t] = (t < K_CLS) ? bm[t] : 0.0f;

  const int lane = t & 31;
  const int w    = t >> 5;       // 8 waves
  const int n0   = w * 16;       // this wave's 16-column slab
  const int half = lane >> 4;    // A/B K-offset select per ISA layout
  const int lm   = lane & 15;

  v8f m0 = {}; v8f m1 = {};      // pre_mask accumulators (M-tiles 0,1)

  // ---- GEMM1: loop 4 channel-chunks of 128, K inside chunk ----
  for (int cc = 0; cc < C_DIM / TS; ++cc) {
    __syncthreads();
    for (int q = t; q < TS * (TS / 4); q += NTHR) {     // stage x chunk (float4)
      const int row = q >> 5;
      const int col = (q & 31) * 4;
      const float4 v = *(const float4*)
          &x[((size_t)(bi * C_DIM + cc * TS + row)) * S_DIM + s0 + col];
      float* d = &lX[row * XPITCH + col];
      d[0] = v.x; d[1] = v.y; d[2] = v.z; d[3] = v.w;
    }
    __syncthreads();
    for (int kk = 0; kk < TS; kk += 4) {
      const int kb = kk + half * 2;
      v2f b;
      b.x = lX[kb * XPITCH + n0 + lm];
      b.y = lX[(kb + 1) * XPITCH + n0 + lm];
      const int gc = cc * TS + kb;
      const v2f a0 = *(const v2f*)&lW[lm * WPITCH + gc];
      const v2f a1 = *(const v2f*)&lW[(16 + lm) * WPITCH + gc];
      m0 = wmma_f32(a0, b, m0);
      m1 = wmma_f32(a1, b, m1);
    }
  }
  __syncthreads();

  // ---- bias + sigmoid, scatter C-layout into lM[32,128] ----
#pragma unroll
  for (int j = 0; j < 8; ++j) {
    const int r0 = j + half * 8;             // C layout: vgpr j -> rows j / j+8
    const float p0 = m0[j] + lB[r0];
    lM[r0 * MPITCH + n0 + lm] = 1.0f / (1.0f + __expf(-p0));
    const int r1 = 16 + r0;
    const float p1 = m1[j] + lB[r1];
    lM[r1 * MPITCH + n0 + lm] = 1.0f / (1.0f + __expf(-p1));
  }

  // ---- GEMM2: cf[32,512] += mask[32,128] . xT[128,512] ----
  for (int cc = 0; cc < C_DIM / TS; ++cc) {
    __syncthreads();
    for (int q = t; q < TS * (TS / 4); q += NTHR) {     // re-stage chunk (L2 hit)
      const int row = q >> 5;
      const int col = (q & 31) * 4;
      const float4 v = *(const float4*)
          &x[((size_t)(bi * C_DIM + cc * TS + row)) * S_DIM + s0 + col];
      float* d = &lX[row * XPITCH + col];
      d[0] = v.x; d[1] = v.y; d[2] = v.z; d[3] = v.w;
    }
    __syncthreads();
    v8f c0 = {}; v8f c1 = {};
    for (int kk = 0; kk < TS; kk += 4) {
      const int ks = kk + half * 2;
      const v2f b  = *(const v2f*)&lX[(n0 + lm) * XPITCH + ks]; // B[K=s][N=c]
      const v2f a0 = *(const v2f*)&lM[lm * MPITCH + ks];
      const v2f a1 = *(const v2f*)&lM[(16 + lm) * MPITCH + ks];
      c0 = wmma_f32(a0, b, c0);
      c1 = wmma_f32(a1, b, c1);
    }
    const int cbase = cc * TS + n0 + lm;
#pragma unroll
    for (int j = 0; j < 8; ++j) {
      const int r0 = j + half * 8;
      if (r0 < K_CLS)
        __hip_atomic_fetch_add(&cf[((size_t)bi * K_CLS + r0) * C_DIM + cbase],
                               c0[j], __ATOMIC_RELAXED, __HIP_MEMORY_SCOPE_AGENT);
      const int r1 = 16 + r0;
      if (r1 < K_CLS)
        __hip_atomic_fetch_add(&cf[((size_t)bi * K_CLS + r1) * C_DIM + cbase],
                               c1[j], __ATOMIC_RELAXED, __HIP_MEMORY_SCOPE_AGENT);
    }
  }
}

// ---------------------------------------------------------------------------
// K2: filters[b,k,o] = sum_i Wf[k,o,i] * (cf[b,k,i]/HW) + bf[k,o]
// grid (8 o-chunks, 19 k); wave handles 8 o's, lanes stride i (coalesced Wf).
// ---------------------------------------------------------------------------
__global__ void __launch_bounds__(NTHR)
k2_filters(const float* __restrict__ Wf, const float* __restrict__ bf,
           const float* __restrict__ cf, float* __restrict__ filt) {
  __shared__ float cfb[B_DIM][C_DIM];
  const int t    = threadIdx.x;
  const int kcls = blockIdx.y;
  const int o0   = blockIdx.x * 64;
  const float INV_HW = 1.0f / (float)S_DIM;

  for (int q = t; q < (B_DIM * C_DIM) / 4; q += NTHR) {
    const int b  = q >> 7;
    const int c4 = (q & 127) * 4;
    const float4 v = *(const float4*)&cf[((size_t)b * K_CLS + kcls) * C_DIM + c4];
    cfb[b][c4 + 0] = v.x * INV_HW; cfb[b][c4 + 1] = v.y * INV_HW;
    cfb[b][c4 + 2] = v.z * INV_HW; cfb[b][c4 + 3] = v.w * INV_HW;
  }
  __syncthreads();

  const int lane = t & 31;
  const int w    = t >> 5;
  for (int oi = 0; oi < 8; ++oi) {
    const int o = o0 + w * 8 + oi;
    const float* wrow = &Wf[((size_t)kcls * C_DIM + o) * C_DIM];
    float acc[B_DIM] = {0.f, 0.f, 0.f, 0.f, 0.f, 0.f, 0.f, 0.f};
    for (int i = lane; i < C_DIM; i += 32) {
      const float wv = wrow[i];
#pragma unroll
      for (int b = 0; b < B_DIM; ++b) acc[b] += wv * cfb[b][i];
    }
#pragma unroll
    for (int b = 0; b < B_DIM; ++b)
      for (int off = 16; off > 0; off >>= 1)
        acc[b] += __shfl_xor(acc[b], off, 32);
    if (lane < B_DIM)
      filt[((size_t)lane * K_CLS + kcls) * C_DIM + o] =
          acc[lane] + bf[kcls * C_DIM + o];
  }
}

// ---------------------------------------------------------------------------
// K3: pred[b,k,s] = sum_c filters[b,k,c] * x[b,c,s]   (pass 2 over x)
// Identical WMMA tiling to K1-GEMM1 with lF in place of lW.
// ---------------------------------------------------------------------------
__global__ void __launch_bounds__(NTHR)
k3_pred(const float* __restrict__ x, const float* __restrict__ filt,
        float* __restrict__ out) {
  extern __shared__ float lds[];
  float* lF = lds;                  // KP * WPITCH
  float* lX = lF + KP * WPITCH;     // TS * XPITCH

  const int t  = threadIdx.x;
  const int bi = blockIdx.y;
  const int s0 = blockIdx.x * TS;

  for (int i = t; i < KP * WPITCH; i += NTHR) lF[i] = 0.0f;
  __syncthreads();
  for (int q = t; q < (K_CLS * C_DIM) / 4; q += NTHR) {
    const int row = q / (C_DIM / 4);
    const int col = (q % (C_DIM / 4)) * 4;
    const float4 v =
        *(const float4*)&filt[((size_t)bi * K_CLS + row) * C_DIM + col];
    float* d = &lF[row * WPITCH + col];
    d[0] = v.x; d[1] = v.y; d[2] = v.z; d[3] = v.w;
  }

  const int lane = t & 31;
  const int w    = t >> 5;
  const int n0   = w * 16;
  const int half = lane >> 4;
  const int lm   = lane & 15;

  v8f p0 = {}; v8f p1 = {};
  for (int cc = 0; cc < C_DIM / TS; ++cc) {
    __syncthreads();
    for (int q = t; q < TS * (TS / 4); q += NTHR) {
      const int row = q >> 5;
      const int col = (q & 31) * 4;
      const float4 v = *(const float4*)
          &x[((size_t)(bi * C_DIM + cc * TS + row)) * S_DIM + s0 + col];
      float* d = &lX[row * XPITCH + col];
      d[0] = v.x; d[1] = v.y; d[2] = v.z; d[3] = v.w;
    }
    __syncthreads();
    for (int kk = 0; kk < TS; kk += 4) {
      const int kb = kk + half * 2;
      v2f b;
      b.x = lX[kb * XPITCH + n0 + lm];
      b.y = lX[(kb + 1) * XPITCH + n0 + lm];
      const int gc = cc * TS + kb;
      const v2f a0 = *(const v2f*)&lF[lm * WPITCH + gc];
      const v2f a1 = *(const v2f*)&lF[(16 + lm) * WPITCH + gc];
      p0 = wmma_f32(a0, b, p0);
      p1 = wmma_f32(a1, b, p1);
    }
  }
#pragma unroll
  for (int j = 0; j < 8; ++j) {
    const int r0 = j + half * 8;
    if (r0 < K_CLS)
      out[((size_t)bi * K_CLS + r0) * S_DIM + s0 + n0 + lm] = p0[j];
    const int r1 = 16 + r0;
    if (r1 < K_CLS)
      out[((size_t)bi * K_CLS + r1) * S_DIM + s0 + n0 + lm] = p1[j];
  }
}

// ---------------------------------------------------------------------------
extern "C" void kernel_launch(void* const* d_in, const int* in_sizes, int n_in,
                              void* d_out, int out_size, void* d_ws,
                              size_t ws_size, hipStream_t stream) {
  (void)in_sizes; (void)n_in; (void)out_size; (void)ws_size;
  const float* x  = (const float*)d_in[0];
  const float* Wm = (const float*)d_in[1];
  const float* bm = (const float*)d_in[2];
  const float* Wf = (const float*)d_in[3];
  const float* bf = (const float*)d_in[4];
  float* out = (float*)d_out;

  float* cf   = (float*)d_ws;                     // [8,19,512]
  float* filt = cf + (size_t)B_DIM * K_CLS * C_DIM;  // [8,19,512]

  hipMemsetAsync(cf, 0, (size_t)B_DIM * K_CLS * C_DIM * sizeof(float), stream);

  const size_t lds1 =
      (size_t)(KP * WPITCH + TS * XPITCH + KP * MPITCH + KP) * sizeof(float);
  k1_mask_pool<<<dim3(S_DIM / TS, B_DIM), NTHR, lds1, stream>>>(x, Wm, bm, cf);

  k2_filters<<<dim3(C_DIM / 64, K_CLS), NTHR, 0, stream>>>(Wf, bf, cf, filt);

  const size_t lds3 = (size_t)(KP * WPITCH + TS * XPITCH) * sizeof(float);
  k3_pred<<<dim3(S_DIM / TS, B_DIM), NTHR, lds3, stream>>>(x, filt, out);
}